// SGATLayer_9577777070578
// MI455X (gfx1250) — compile-verified
//
#include <hip/hip_runtime.h>
#include <hip/hip_bf16.h>

#define NN 100000
#define EE 1600000
#define IN_CH 128
#define ED 32
#define H 4
#define C 32
#define HC 128
#define NEG_SLOPE 0.2f

typedef float v2f __attribute__((ext_vector_type(2)));
typedef float v8f __attribute__((ext_vector_type(8)));

// ---------- order-preserving float <-> uint key for atomic max ----------
__device__ __forceinline__ unsigned fkey(float f) {
    unsigned b = __float_as_uint(f);
    return (b & 0x80000000u) ? ~b : (b | 0x80000000u);
}
__device__ __forceinline__ float funkey(unsigned k) {
    unsigned b = (k & 0x80000000u) ? (k ^ 0x80000000u) : ~k;
    return __uint_as_float(b);
}

// ---------- Kernel 1: xh = x @ W via V_WMMA_F32_16X16X4_F32 ----------
// grid.x = N/16 = 6250 row tiles; 8 waves/block, wave w owns 16-col tile w.
__global__ __launch_bounds__(256) void gemm_xh_wmma(const float* __restrict__ x,
                                                    const float* __restrict__ W,
                                                    float* __restrict__ xh) {
    const int wave = threadIdx.x >> 5;          // 0..7 -> col tile
    const int lane = threadIdx.x & 31;
    const int half = lane >> 4;                 // 0: lanes 0-15, 1: lanes 16-31
    const int m    = lane & 15;
    const int row0 = blockIdx.x * 16;
    const int col0 = wave * 16;

    v8f acc = {};
    const float* __restrict__ xrow = x + (size_t)(row0 + m) * IN_CH;
    #pragma unroll 4
    for (int k = 0; k < IN_CH; k += 4) {
        const int ka = k + 2 * half;
        v2f a, b;
        // A 16x4 fp32 layout: vgpr j, lane -> A[m][2*half + j]
        a.x = xrow[ka];
        a.y = xrow[ka + 1];
        // B 4x16 fp32 layout: vgpr j, lane -> B[2*half + j][m]
        b.x = W[(size_t)ka * HC + col0 + m];
        b.y = W[(size_t)(ka + 1) * HC + col0 + m];
        acc = __builtin_amdgcn_wmma_f32_16x16x4_f32(false, a, false, b,
                                                    (short)0, acc, false, false);
    }
    // D 16x16 fp32 layout: vgpr r, lane -> D[r + 8*half][m]
    #pragma unroll
    for (int r = 0; r < 8; ++r)
        xh[(size_t)(row0 + r + 8 * half) * HC + col0 + m] = acc[r];
}

// ---------- Kernel 2: per-node attention dots a_src/a_dst ----------
// 256 threads = 2 nodes x 128 channels; each wave (32 lanes) == one head.
__global__ __launch_bounds__(256) void node_attn(const float* __restrict__ xh,
                                                 const float* __restrict__ att_src,
                                                 const float* __restrict__ att_dst,
                                                 float* __restrict__ a_src,
                                                 float* __restrict__ a_dst) {
    const int t = threadIdx.x & 127;                       // channel 0..127
    const int n = blockIdx.x * 2 + (threadIdx.x >> 7);
    if (n >= NN) return;
    const float v  = xh[(size_t)n * HC + t];
    float ps = v * att_src[t];
    float pd = v * att_dst[t];
    #pragma unroll
    for (int off = 16; off > 0; off >>= 1) {
        ps += __shfl_down(ps, off, 32);
        pd += __shfl_down(pd, off, 32);
    }
    if ((t & 31) == 0) {
        a_src[n * H + (t >> 5)] = ps;
        a_dst[n * H + (t >> 5)] = pd;
    }
}

// ---------- Kernel 3: v[d][h] = sum_c W_edge[d][h*32+c] * att_edge[h][c] ----------
__global__ void edge_vec(const float* __restrict__ W_edge,
                         const float* __restrict__ att_edge,
                         float* __restrict__ v) {
    const int t = threadIdx.x;            // 0..127
    const int d = t >> 2, h = t & 3;
    float s = 0.f;
    #pragma unroll 8
    for (int c = 0; c < C; ++c)
        s += W_edge[d * HC + h * C + c] * att_edge[h * C + c];
    v[d * H + h] = s;
}

// ---------- Kernel 4: edge logits + leaky relu + segment max (atomic key max) ----------
__global__ __launch_bounds__(256) void edge_logits_kernel(const float* __restrict__ edge_attr,
                                                          const int* __restrict__ src,
                                                          const int* __restrict__ dst,
                                                          const float* __restrict__ a_src,
                                                          const float* __restrict__ a_dst,
                                                          const float* __restrict__ v,
                                                          float* __restrict__ logits,
                                                          unsigned* __restrict__ segmax) {
    const int e = blockIdx.x * blockDim.x + threadIdx.x;
    if (e >= EE) return;
    const float* __restrict__ ea = edge_attr + (size_t)e * ED;
    float acc[H] = {0.f, 0.f, 0.f, 0.f};
    #pragma unroll 4
    for (int d = 0; d < ED; ++d) {
        const float a = ea[d];
        #pragma unroll
        for (int h = 0; h < H; ++h) acc[h] += a * v[d * H + h];
    }
    const int si = src[e], di = dst[e];
    #pragma unroll
    for (int h = 0; h < H; ++h) {
        float lg = a_src[si * H + h] + a_dst[di * H + h] + acc[h];
        lg = lg >= 0.f ? lg : NEG_SLOPE * lg;
        logits[(size_t)e * H + h] = lg;
        atomicMax(&segmax[di * H + h], fkey(lg));
    }
}

// ---------- Kernel 5: expv = exp(logit - segmax[dst]); denom += expv ----------
__global__ __launch_bounds__(256) void edge_exp_kernel(const int* __restrict__ dst,
                                                       const unsigned* __restrict__ segmax,
                                                       float* __restrict__ logits,   // in: logits, out: expv
                                                       float* __restrict__ denom) {
    const long gid = (long)blockIdx.x * blockDim.x + threadIdx.x;  // e*H + h
    if (gid >= (long)EE * H) return;
    const int e = (int)(gid >> 2), h = (int)(gid & 3);
    const int di = dst[e];
    const float mx = funkey(segmax[di * H + h]);
    const float ev = expf(logits[gid] - mx);
    logits[gid] = ev;
    atomicAdd(&denom[di * H + h], ev);
}

// ---------- Kernel 6: scatter messages, one wave per edge ----------
__global__ __launch_bounds__(256) void scatter_msg(const int* __restrict__ src,
                                                   const int* __restrict__ dst,
                                                   const float* __restrict__ expv,
                                                   const float* __restrict__ denom,
                                                   const float* __restrict__ xh,
                                                   float* __restrict__ accum) {
    const int e = (blockIdx.x * blockDim.x + threadIdx.x) >> 5;
    const int lane = threadIdx.x & 31;
    if (e >= EE) return;
    const int si = src[e], di = dst[e];
    const float* __restrict__ xrow = xh + (size_t)si * HC;
    float* __restrict__ orow = accum + (size_t)di * HC;
    #pragma unroll
    for (int h = 0; h < H; ++h) {
        const float alpha = expv[(size_t)e * H + h] / denom[di * H + h];
        const int t = h * 32 + lane;
        atomicAdd(&orow[t], xrow[t] * alpha);
    }
}

// ---------- Kernel 7: finalize: mean over heads + biases + relu ----------
__global__ __launch_bounds__(256) void finalize(const float* __restrict__ accum,
                                                const float* __restrict__ gat_bias,
                                                const float* __restrict__ bias,
                                                float* __restrict__ out) {
    const int gid = blockIdx.x * blockDim.x + threadIdx.x;  // n*C + c
    if (gid >= NN * C) return;
    const int n = gid >> 5, c = gid & 31;
    float s = 0.f;
    #pragma unroll
    for (int h = 0; h < H; ++h)
        s += accum[(size_t)n * HC + h * C + c] + gat_bias[h * C + c];
    s *= 0.25f;
    s += bias[c];
    out[gid] = s > 0.f ? s : 0.f;
}

static inline size_t align_up(size_t v, size_t a) { return (v + a - 1) & ~(a - 1); }

extern "C" void kernel_launch(void* const* d_in, const int* in_sizes, int n_in,
                              void* d_out, int out_size, void* d_ws, size_t ws_size,
                              hipStream_t stream) {
    const float* x         = (const float*)d_in[0];
    const int*   edge_idx  = (const int*)  d_in[1];   // (2, E) flat
    const float* edge_attr = (const float*)d_in[2];
    const float* W         = (const float*)d_in[3];
    const float* att_src   = (const float*)d_in[4];
    const float* att_dst   = (const float*)d_in[5];
    const float* W_edge    = (const float*)d_in[6];
    const float* att_edge  = (const float*)d_in[7];
    const float* gat_bias  = (const float*)d_in[8];
    const float* bias      = (const float*)d_in[9];
    float* out = (float*)d_out;

    const int* src = edge_idx;
    const int* dst = edge_idx + EE;

    // workspace layout
    char* base = (char*)d_ws;
    size_t off = 0;
    float* xh      = (float*)(base + off); off = align_up(off + (size_t)NN * HC * 4, 256);
    float* logits  = (float*)(base + off); off = align_up(off + (size_t)EE * H  * 4, 256);
    float* a_src   = (float*)(base + off); off = align_up(off + (size_t)NN * H  * 4, 256);
    float* a_dst   = (float*)(base + off); off = align_up(off + (size_t)NN * H  * 4, 256);
    float* denom   = (float*)(base + off); off = align_up(off + (size_t)NN * H  * 4, 256);
    unsigned* smax = (unsigned*)(base + off); off = align_up(off + (size_t)NN * H * 4, 256);
    float* vvec    = (float*)(base + off); off = align_up(off + (size_t)ED * H  * 4, 256);
    float* accum   = (float*)(base + off); off = align_up(off + (size_t)NN * HC * 4, 256);
    (void)ws_size;

    // init accumulators (capture-safe async memsets)
    hipMemsetAsync(smax,  0, (size_t)NN * H  * 4, stream);
    hipMemsetAsync(denom, 0, (size_t)NN * H  * 4, stream);
    hipMemsetAsync(accum, 0, (size_t)NN * HC * 4, stream);

    // 1) xh = x @ W   (WMMA fp32)
    gemm_xh_wmma<<<NN / 16, 256, 0, stream>>>(x, W, xh);

    // 2) a_src / a_dst
    node_attn<<<(NN + 1) / 2, 256, 0, stream>>>(xh, att_src, att_dst, a_src, a_dst);

    // 3) folded edge-attention vector v = reduce(W_edge, att_edge)
    edge_vec<<<1, 128, 0, stream>>>(W_edge, att_edge, vvec);

    // 4) logits + leaky relu + segment max
    edge_logits_kernel<<<(EE + 255) / 256, 256, 0, stream>>>(edge_attr, src, dst,
                                                             a_src, a_dst, vvec,
                                                             logits, smax);

    // 5) exp + denominator
    edge_exp_kernel<<<((long)EE * H + 255) / 256, 256, 0, stream>>>(dst, smax, logits, denom);

    // 6) weighted scatter of messages (wave per edge)
    scatter_msg<<<(EE * 32 + 255) / 256, 256, 0, stream>>>(src, dst, logits, denom, xh, accum);

    // 7) head-mean + biases + relu
    finalize<<<(NN * C + 255) / 256, 256, 0, stream>>>(accum, gat_bias, bias, out);
}